// MVGlimpseNetwork_71098888618099
// MI455X (gfx1250) — compile-verified
//
#include <hip/hip_runtime.h>
#include <hip/hip_bf16.h>

// Problem constants (match reference)
#define BB    128
#define TT    4096
#define VV    32
#define QQ    64      // NGRAN * NGLIMPSE
#define GC    2048    // VV * QQ
#define NHID  256
#define FANIN 2049

typedef __attribute__((ext_vector_type(2))) float v2f;
typedef __attribute__((ext_vector_type(8))) float v8f;

// ---------------------------------------------------------------------------
// Kernel 0: LDS-tiled transpose W[n][k] (n<256, k<2048) -> Wt[k][n].
// Reads are contiguous along k (W rows), writes contiguous along n.
// ---------------------------------------------------------------------------
__global__ __launch_bounds__(256) void k_transpose_w(const float* __restrict__ W,
                                                     float* __restrict__ Wt) {
    __shared__ float tile[32][33];          // +1 pitch: no bank conflicts
    const int k0 = blockIdx.x * 32;         // 64 blocks
    const int n0 = blockIdx.y * 32;         // 8 blocks
    const int tx = threadIdx.x & 31;
    const int ty = threadIdx.x >> 5;        // 0..7
#pragma unroll
    for (int j = 0; j < 4; ++j) {
        const int nl = ty + 8 * j;
        tile[nl][tx] = W[(size_t)(n0 + nl) * FANIN + (k0 + tx)];   // coalesced read
    }
    __syncthreads();
#pragma unroll
    for (int j = 0; j < 4; ++j) {
        const int kl = ty + 8 * j;
        Wt[(size_t)(k0 + kl) * NHID + (n0 + tx)] = tile[tx][kl];   // coalesced write
    }
}

// ---------------------------------------------------------------------------
// Kernel 1: glimpse interpolation. One workgroup per batch b.
//  - LDS: time row (4096 f32) + validity bitset (4096 u32 words, bit v of
//    word t == masks[b][t][v] != 0) + per-v tmax/any.
//  - Each query: binary search the (already sorted) time row, then scan the
//    bitset backward/forward for the bracketing valid samples, interpolate
//    with endpoint clamping exactly like searchsorted + clip in the reference.
//  Only `masks` is read in full (64 MB); `vals` is touched sparsely.
// ---------------------------------------------------------------------------
__global__ __launch_bounds__(256) void k_glimpse(const float* __restrict__ vals,
                                                 const float* __restrict__ time,
                                                 const int*   __restrict__ masks,
                                                 const float* __restrict__ l_t,
                                                 float* __restrict__ g,
                                                 float* __restrict__ gmid) {
    __shared__ float    s_time[TT];
    __shared__ unsigned s_word[TT];
    __shared__ float    s_tmax[VV];
    __shared__ int      s_any[VV];

    const int b   = blockIdx.x;
    const int tid = threadIdx.x;

    // stage time row
    const float* trow = time + (size_t)b * TT;
    for (int t = tid; t < TT; t += 256) s_time[t] = trow[t];

    // build validity bitset: each thread owns whole 128B mask rows (int4 x8)
    const int* mrow = masks + (size_t)b * TT * VV;
    for (int t = tid; t < TT; t += 256) {
        const int4* mp = (const int4*)(mrow + (size_t)t * VV);
        unsigned w = 0u;
#pragma unroll
        for (int c = 0; c < 8; ++c) {
            int4 m4 = mp[c];
            w |= (m4.x != 0 ? 1u : 0u) << (4 * c + 0);
            w |= (m4.y != 0 ? 1u : 0u) << (4 * c + 1);
            w |= (m4.z != 0 ? 1u : 0u) << (4 * c + 2);
            w |= (m4.w != 0 ? 1u : 0u) << (4 * c + 3);
        }
        s_word[t] = w;
    }
    __syncthreads();

    // per-variable tmax = time of last valid sample (expected ~2 probes)
    if (tid < VV) {
        int v = tid, last = -1;
        for (int t = TT - 1; t >= 0; --t) {
            if ((s_word[t] >> v) & 1u) { last = t; break; }
        }
        s_any[v]  = (last >= 0);
        s_tmax[v] = (last >= 0) ? s_time[last] : 0.0f;
    }
    __syncthreads();

    const float  lt   = l_t[b];
    const float* vrow = vals + (size_t)b * TT * VV;

    for (int qid = tid; qid < GC; qid += 256) {
        const int v  = qid >> 6;          // variable
        const int gi = (qid >> 5) & 1;    // granularity (0: i=1, 1: i=5)
        const int k  = qid & 31;          // glimpse index
        float out = 0.0f;
        if (s_any[v]) {
            const float hw  = gi ? 0.625f : 0.125f;          // GWIDTH*i/2
            const float off = -hw + (float)k * (2.0f * hw / 31.0f);
            const float q   = (off + lt) * s_tmax[v];

            // lower_bound over the full (sorted) time row
            int lo = 0, hi = TT;
            while (lo < hi) {
                int mid = (lo + hi) >> 1;
                if (s_time[mid] < q) lo = mid + 1; else hi = mid;
            }
            const int p = lo;

            // last valid strictly before q
            int il = -1;
            for (int t = p - 1; t >= 0; --t)
                if ((s_word[t] >> v) & 1u) { il = t; break; }
            // first valid at/after q
            int ir = -1;
            for (int t = p; t < TT; ++t)
                if ((s_word[t] >> v) & 1u) { ir = t; break; }

            if (il < 0) {                                    // clamp to first valid
                out = vrow[(size_t)ir * VV + v];
            } else if (ir < 0) {                             // clamp to last valid
                out = vrow[(size_t)il * VV + v];
            } else {
                const float tl = s_time[il], tr = s_time[ir];
                const float vl = vrow[(size_t)il * VV + v];
                const float vr = vrow[(size_t)ir * VV + v];
                const float denom = tr - tl;
                out = (denom > 0.0f) ? vl + (q - tl) / denom * (vr - vl) : vl;
            }
        }
        g[(size_t)b * GC + qid] = out;
        if (qid == GC / 2) gmid[b] = out;   // reference's g[:, g.shape[1]//2]
    }
}

// ---------------------------------------------------------------------------
// Kernel 2: grep = [g | l_t] @ W.T + b  via V_WMMA_F32_16X16X4_F32.
// 128 waves, one 16x16 tile each: 8 M-tiles (over B=128) x 16 N-tiles (NHID).
// A (16x4 f32):  lane-half h, component j -> K = 2h + j, M = lane&15
// B (4x16 f32):  component j, lane-half h -> K = 2h + j, N = lane&15
// C (16x16 f32): VGPR r -> row r (lanes 0-15) / r+8 (lanes 16-31)
// Software-pipelined with two accumulators so each v_wmma depends only on
// loads issued one iteration earlier. l_t * W[:,2048] + bias in the epilogue.
// ---------------------------------------------------------------------------
__global__ __launch_bounds__(256) void k_gemm(const float* __restrict__ g,
                                              const float* __restrict__ Wt,
                                              const float* __restrict__ W,
                                              const float* __restrict__ bias,
                                              const float* __restrict__ l_t,
                                              float* __restrict__ out) {
    const int wave  = (blockIdx.x * blockDim.x + threadIdx.x) >> 5;  // 0..127
    const int lane  = threadIdx.x & 31;
    const int mtile = wave >> 4;        // 0..7
    const int ntile = wave & 15;        // 0..15
    const int mbase = mtile * 16;
    const int nbase = ntile * 16;
    const int half  = lane >> 4;        // 0 or 1
    const int l16   = lane & 15;

    // per-lane base pointers
    const float* arow = g  + (size_t)(mbase + l16) * GC + 2 * half;   // A stream
    const float* bcol = Wt + (size_t)(2 * half) * NHID + nbase + l16; // B stream

#define LD_A(k) (*(const v2f*)(arow + (k)))
#define LD_B(k) (v2f{bcol[(size_t)(k) * NHID], bcol[(size_t)(k) * NHID + NHID]})

    v8f c0 = {}, c1 = {};
    v2f a0 = LD_A(0), b0 = LD_B(0);
    v2f a1 = LD_A(4), b1 = LD_B(4);

    for (int k0 = 8; k0 < GC; k0 += 8) {
        // prefetch B stream well ahead (emits global_prefetch)
        if (k0 + 128 < GC)
            __builtin_prefetch(bcol + (size_t)(k0 + 128) * NHID, 0, 1);
        c0 = __builtin_amdgcn_wmma_f32_16x16x4_f32(false, a0, false, b0,
                                                   (short)0, c0, false, false);
        a0 = LD_A(k0);
        b0 = LD_B(k0);
        c1 = __builtin_amdgcn_wmma_f32_16x16x4_f32(false, a1, false, b1,
                                                   (short)0, c1, false, false);
        a1 = LD_A(k0 + 4);
        b1 = LD_B(k0 + 4);
    }
    c0 = __builtin_amdgcn_wmma_f32_16x16x4_f32(false, a0, false, b0,
                                               (short)0, c0, false, false);
    c1 = __builtin_amdgcn_wmma_f32_16x16x4_f32(false, a1, false, b1,
                                               (short)0, c1, false, false);
    v8f c = c0 + c1;
#undef LD_A
#undef LD_B

    // epilogue: + l_t[m] * W[n][2048] + bias[n]
    const int   n  = nbase + l16;
    const float wl = W[(size_t)n * FANIN + (FANIN - 1)];
    const float bs = bias[n];
#pragma unroll
    for (int r = 0; r < 8; ++r) {
        const int m = mbase + r + half * 8;
        out[(size_t)m * NHID + n] = c[r] + l_t[m] * wl + bs;
    }
}

// ---------------------------------------------------------------------------
// Launch
// ---------------------------------------------------------------------------
extern "C" void kernel_launch(void* const* d_in, const int* in_sizes, int n_in,
                              void* d_out, int out_size, void* d_ws, size_t ws_size,
                              hipStream_t stream) {
    const float* vals  = (const float*)d_in[0];   // [B,T,V] f32
    const float* time  = (const float*)d_in[1];   // [B,T]   f32
    const int*   masks = (const int*)  d_in[2];   // [B,T,V] i32
    // d_in[3] = lengths (unused)
    const float* l_t   = (const float*)d_in[4];   // [B,1]   f32
    const float* W     = (const float*)d_in[5];   // [256,2049] f32
    const float* bias  = (const float*)d_in[6];   // [256]   f32

    float* grep = (float*)d_out;                  // [B,256]
    float* gmid = grep + (size_t)BB * NHID;       // [B]

    float* Wt = (float*)d_ws;                     // [2048,256]  (2 MB)
    float* g  = Wt + (size_t)GC * NHID;           // [B,2048]    (1 MB)

    k_transpose_w<<<dim3(GC / 32, NHID / 32), 256, 0, stream>>>(W, Wt);
    k_glimpse<<<BB, 256, 0, stream>>>(vals, time, masks, l_t, g, gmid);
    k_gemm<<<(BB / 16) * (NHID / 16) / 8 /* =16 blocks of 8 waves */, 256, 0, stream>>>(
        g, Wt, W, bias, l_t, grep);
}